// AvgPool_68934225101080
// MI455X (gfx1250) — compile-verified
//
#include <hip/hip_runtime.h>

// Problem constants (from reference)
constexpr int BATCH = 8;
constexpr int NPTS  = 8192;
constexpr int MOUT  = 2048;
constexpr int KNN   = 32;
constexpr int CHAN  = 128;

typedef __attribute__((ext_vector_type(2))) float v2f;
typedef __attribute__((ext_vector_type(8))) float v8f;

// ---------------------------------------------------------------------------
// Kernel 1: farthest point sampling. One block per batch, 1024 threads,
// 8 points per thread held entirely in registers (coords + running dist).
// Squared distances (order-equivalent to the reference's L2 norms).
// Writes out_coordinates (B, M, 3) directly into d_out.
// ---------------------------------------------------------------------------
__global__ __launch_bounds__(1024) void fps_kernel(const float* __restrict__ coords,
                                                   float* __restrict__ outc) {
  const int b    = blockIdx.x;
  const int tid  = threadIdx.x;
  const int lane = tid & 31;
  const int w    = tid >> 5;   // 32 waves

  const float* cb = coords + (size_t)b * NPTS * 3;

  float x[8], y[8], z[8], dist[8];
#pragma unroll
  for (int j = 0; j < 8; ++j) {
    int p = j * 1024 + tid;            // strided ownership
    x[j] = cb[p * 3 + 0];
    y[j] = cb[p * 3 + 1];
    z[j] = cb[p * 3 + 2];
    dist[j] = 1e20f;
  }

  __shared__ float s_cx, s_cy, s_cz;
  __shared__ float redV[32];
  __shared__ int   redI[32];
  __shared__ int   sWin;

  int winner = 0;  // reference: first sampled index is 0

  for (int i = 0; i < MOUT; ++i) {
    // Owner of `winner` publishes centroid + output coordinate for slot i.
    int ot = winner & 1023;
    int oj = winner >> 10;
    if (tid == ot) {
      s_cx = x[oj]; s_cy = y[oj]; s_cz = z[oj];
      float* o = outc + ((size_t)b * MOUT + i) * 3;
      o[0] = x[oj]; o[1] = y[oj]; o[2] = z[oj];
    }
    __syncthreads();
    if (i == MOUT - 1) break;

    const float lx = s_cx, ly = s_cy, lz = s_cz;
    float bv = -1.0f;
    int   bi = 0x7fffffff;
#pragma unroll
    for (int j = 0; j < 8; ++j) {
      float dx = x[j] - lx, dy = y[j] - ly, dz = z[j] - lz;
      float d = dx * dx + dy * dy + dz * dz;
      d = fminf(dist[j], d);
      dist[j] = d;
      int p = j * 1024 + tid;
      if (d > bv || (d == bv && p < bi)) { bv = d; bi = p; }
    }
    // wave-level argmax (tie -> lowest index, matching jnp.argmax)
#pragma unroll
    for (int off = 16; off >= 1; off >>= 1) {
      float ov = __shfl_xor(bv, off);
      int   oi = __shfl_xor(bi, off);
      if (ov > bv || (ov == bv && oi < bi)) { bv = ov; bi = oi; }
    }
    if (lane == 0) { redV[w] = bv; redI[w] = bi; }
    __syncthreads();
    if (w == 0) {
      bv = redV[lane];
      bi = redI[lane];
#pragma unroll
      for (int off = 16; off >= 1; off >>= 1) {
        float ov = __shfl_xor(bv, off);
        int   oi = __shfl_xor(bi, off);
        if (ov > bv || (ov == bv && oi < bi)) { bv = ov; bi = oi; }
      }
      if (lane == 0) sWin = bi;
    }
    __syncthreads();
    winner = sWin;
  }
}

// ---------------------------------------------------------------------------
// Kernel 2: fused KNN (WMMA f32 16x16x4 distance matrix + wave32 bitonic
// top-32) and neighbor-feature mean.
// Grid: (M/16 tiles, B). Block: 512 threads = 16 waves. Wave w owns center w.
// A row m = (-2cx, -2cy, -2cz, 1);  B col n = (px, py, pz, p^2)
//  => D[m][n] = p^2 - 2 c.p = d^2 - c^2   (c^2 is per-row constant: top-k safe)
// ---------------------------------------------------------------------------
constexpr int CHUNK = 512;       // points per chunk
constexpr int ROWP  = 516;       // padded LDS row stride (floats)

__global__ __launch_bounds__(512) void knn_mean_kernel(const float* __restrict__ coords,
                                                       const float* __restrict__ feats,
                                                       const float* __restrict__ outc,
                                                       float* __restrict__ outf) {
  const int tile = blockIdx.x;     // 0..127 (16 centers each)
  const int b    = blockIdx.y;     // 0..7
  const int tid  = threadIdx.x;
  const int lane = tid & 31;
  const int w    = tid >> 5;       // 0..15

  __shared__ float ds[16 * ROWP];        // 16 centers x 512-pt distance chunk
  __shared__ int   nbr[16][KNN];

  const float* cb = coords + (size_t)b * NPTS * 3;

  // --- A fragment: 16 FPS centers (identical in every wave) ---
  // 16x4 f32 A layout: lanes 0-15 hold K=0 (v0) / K=1 (v1) for M=lane;
  //                    lanes 16-31 hold K=2 (v0) / K=3 (v1) for M=lane-16.
  const int ci = lane & 15;
  const float* cc = outc + ((size_t)b * MOUT + tile * 16 + ci) * 3;
  const float ccx = cc[0], ccy = cc[1], ccz = cc[2];
  v2f afrag;
  afrag[0] = (lane < 16) ? (-2.0f * ccx) : (-2.0f * ccz);
  afrag[1] = (lane < 16) ? (-2.0f * ccy) : 1.0f;

  // running top-32 (ascending across lanes; lane 31 holds current max)
  float topd = 1e30f;
  int   topi = 0;

  for (int chunk = 0; chunk < NPTS / CHUNK; ++chunk) {
    // ---- Phase A: 32 WMMA point-tiles per chunk, 2 per wave ----
#pragma unroll
    for (int tt = 0; tt < 2; ++tt) {
      const int ptile = w + tt * 16;                       // 0..31
      const int pi    = chunk * CHUNK + ptile * 16 + (lane & 15);
      const float px = cb[pi * 3 + 0];
      const float py = cb[pi * 3 + 1];
      const float pz = cb[pi * 3 + 2];
      const float p2 = px * px + py * py + pz * pz;
      // 4x16 f32 B layout: lanes 0-15 hold K=0 (v0) / K=1 (v1) for N=lane;
      //                    lanes 16-31 hold K=2 (v0) / K=3 (v1) for N=lane-16.
      v2f bfrag;
      bfrag[0] = (lane < 16) ? px : pz;
      bfrag[1] = (lane < 16) ? py : p2;

      v8f acc = {0.f, 0.f, 0.f, 0.f, 0.f, 0.f, 0.f, 0.f};
      acc = __builtin_amdgcn_wmma_f32_16x16x4_f32(
          /*neg_a=*/false, afrag, /*neg_b=*/false, bfrag,
          /*c_mod=*/(short)0, acc, /*reuse_a=*/false, /*reuse_b=*/false);

      // C/D layout: vgpr r: lanes 0-15 -> (M=r, N=lane); lanes 16-31 -> (M=r+8)
      const int colc  = ptile * 16 + (lane & 15);
      const int mbase = (lane < 16) ? 0 : 8;
#pragma unroll
      for (int r = 0; r < 8; ++r)
        ds[(mbase + r) * ROWP + colc] = acc[r];
    }
    __syncthreads();

    // ---- Phase B: wave w merges its center's 512 distances into top-32 ----
#pragma unroll 4
    for (int g = 0; g < CHUNK / 32; ++g) {
      float cand = ds[w * ROWP + g * 32 + lane];
      int   cidx = chunk * CHUNK + g * 32 + lane;
      const float curmax = __shfl(topd, 31);
      if (__ballot(cand < curmax)) {            // wave-uniform skip
        // bitonic sort the 32 candidates ascending (index tiebreak)
        float ck = cand; int cv = cidx;
        for (int size = 2; size <= 32; size <<= 1) {
          for (int stride = size >> 1; stride >= 1; stride >>= 1) {
            float ok = __shfl_xor(ck, stride);
            int   ov = __shfl_xor(cv, stride);
            bool keepmin = (((lane & size) == 0) == ((lane & stride) == 0));
            bool take = keepmin ? (ok < ck || (ok == ck && ov < cv))
                                : (ok > ck || (ok == ck && ov > cv));
            if (take) { ck = ok; cv = ov; }
          }
        }
        // reverse -> descending, elementwise min with ascending top list:
        // yields the union's 32 smallest as a bitonic sequence
        float rk = __shfl(ck, 31 - lane);
        int   rv = __shfl(cv, 31 - lane);
        if (rk < topd || (rk == topd && rv < topi)) { topd = rk; topi = rv; }
        // 5-stage bitonic merge back to ascending
#pragma unroll
        for (int stride = 16; stride >= 1; stride >>= 1) {
          float ok = __shfl_xor(topd, stride);
          int   ov = __shfl_xor(topi, stride);
          bool keepmin = ((lane & stride) == 0);
          bool take = keepmin ? (ok < topd || (ok == topd && ov < topi))
                              : (ok > topd || (ok == topd && ov > topi));
          if (take) { topd = ok; topi = ov; }
        }
      }
    }
    __syncthreads();   // protect ds before next chunk overwrites it
  }

  nbr[w][lane] = topi;
  __syncthreads();

  // ---- Phase C: neighbor-feature mean, coalesced along channels ----
  const float* fb = feats + (size_t)b * NPTS * CHAN;
  const int ch = tid & (CHAN - 1);   // 0..127
  const int mg = tid >> 7;           // 0..3
#pragma unroll
  for (int gi = 0; gi < 4; ++gi) {
    const int m = gi * 4 + mg;
    float acc = 0.0f;
#pragma unroll
    for (int k = 0; k < KNN; ++k)
      acc += fb[(size_t)nbr[m][k] * CHAN + ch];
    outf[((size_t)b * MOUT + tile * 16 + m) * CHAN + ch] = acc * (1.0f / 32.0f);
  }
}

// ---------------------------------------------------------------------------
extern "C" void kernel_launch(void* const* d_in, const int* in_sizes, int n_in,
                              void* d_out, int out_size, void* d_ws, size_t ws_size,
                              hipStream_t stream) {
  (void)in_sizes; (void)n_in; (void)out_size; (void)d_ws; (void)ws_size;
  const float* coords = (const float*)d_in[0];   // (8, 8192, 3)   f32
  const float* feats  = (const float*)d_in[1];   // (8, 8192, 128) f32
  float* outc = (float*)d_out;                           // (8, 2048, 3)
  float* outf = outc + (size_t)BATCH * MOUT * 3;         // (8, 2048, 128)

  fps_kernel<<<dim3(BATCH), dim3(1024), 0, stream>>>(coords, outc);
  knn_mean_kernel<<<dim3(MOUT / 16, BATCH), dim3(512), 0, stream>>>(coords, feats, outc, outf);
}